// SSMKernel_69432441307354
// MI455X (gfx1250) — compile-verified
//
#include <hip/hip_runtime.h>

// ---------------------------------------------------------------------------
// Mamba/SSD chunked selective scan for MI455X (gfx1250, wave32, WMMA + TDM).
//   B=2, L=2048, NH=24, DH=64, DS=128.  One workgroup per (b,h) stream.
//   Chunk T=64: serial chain 2048 -> 32 steps; chunk work is f32 WMMA
//   16x16x4 matmuls; B/C tiles are double-buffered via tensor_load_to_lds
//   (TDM does the +4-dword row padding in hardware).
// ---------------------------------------------------------------------------

typedef __attribute__((ext_vector_type(2))) float v2f;
typedef __attribute__((ext_vector_type(8))) float v8f;
typedef __attribute__((ext_vector_type(4))) unsigned int u32x4;
typedef __attribute__((ext_vector_type(8))) int i32x8;
typedef __attribute__((ext_vector_type(4))) int i32x4;

#define BB 2
#define LL 2048
#define NHEAD 24
#define DHD 64
#define DST 128
#define CT 64              // chunk length
#define NC (LL / CT)       // 32 chunks
#define NT 256             // 8 wave32 waves

// padded LDS row strides (lane-to-lane bank delta 4 on transposed frag loads)
#define XP 68              // sX [CT][XP]
#define BP 132             // sB [CT][BP]   (TDM pad: 128 dw + 4 dw)
#define CP 132             // sC [CT][CP]
#define HP 132             // sH [DHD][HP]
#define SP 68              // sS [CT][SP]

#define SZ_X (CT * XP)     // 4352
#define SZ_BC (CT * BP)    // 8448
#define SZ_H (DHD * HP)    // 8448
#define OFF_X 0
#define OFF_B0 (OFF_X + SZ_X)
#define OFF_B1 (OFF_B0 + SZ_BC)
#define OFF_C0 (OFF_B1 + SZ_BC)
#define OFF_C1 (OFF_C0 + SZ_BC)
#define OFF_H  (OFF_C1 + SZ_BC)
#define OFF_S  (OFF_H + SZ_H)
#define OFF_LRAW (OFF_S + CT * SP)
#define OFF_LG   (OFF_LRAW + CT)
#define OFF_E    (OFF_LG + CT)
#define OFF_DT   (OFF_E + CT)
#define LDS_FLOATS (OFF_DT + CT)   // 51200 floats = 204800 B (< 320KB WGP LDS)

#if defined(__has_builtin)
#if __has_builtin(__builtin_amdgcn_tensor_load_to_lds) && \
    __has_builtin(__builtin_amdgcn_s_wait_tensorcnt)
#define HAVE_TDM 1
#endif
#endif

// --- fragment loaders (CDNA5 ISA 16x4 f32 A layout) -------------------
__device__ __forceinline__ v2f ld_a(const float* p, int stride) {
  int lane = threadIdx.x & 31;
  int m = lane & 15;
  int k0 = (lane >> 4) << 1;
  v2f a;
  a.x = p[m * stride + k0];
  a.y = p[m * stride + k0 + 1];
  return a;
}
__device__ __forceinline__ v2f ld_b(const float* p, int stride) {
  int lane = threadIdx.x & 31;
  int n = lane & 15;
  int k0 = (lane >> 4) << 1;
  v2f b;
  b.x = p[k0 * stride + n];
  b.y = p[(k0 + 1) * stride + n];
  return b;
}
__device__ __forceinline__ v8f wmma_f32(v2f a, v2f b, v8f c) {
  return __builtin_amdgcn_wmma_f32_16x16x4_f32(false, a, false, b, (short)0, c,
                                               false, false);
}

#ifdef HAVE_TDM
// TDM: load a CT x DST f32 tile (row stride NHEAD*DST elems) into LDS with
// 4-dword padding after every 128 dwords (-> BP=132 row stride).  D# built
// per CDNA5 ISA.  Issued by one wave.
__device__ __forceinline__ void tdm_load_tile(const float* gsrc,
                                              unsigned lds_byte_off) {
  unsigned long long ga = (unsigned long long)(uintptr_t)(const void*)gsrc;
  u32x4 g0;
  g0[0] = 1u;                                        // count=1, user mode
  g0[1] = lds_byte_off;                              // lds_addr (bytes)
  g0[2] = (unsigned)(ga & 0xffffffffu);              // global_addr[31:0]
  g0[3] = (unsigned)((ga >> 32) & 0x01ffffffu)       // global_addr[56:32]
          | (2u << 30);                              // type = 2 ("image")
  i32x8 g1;
  g1[0] = (int)((2u << 16)        // data_size = 4B
                | (1u << 20)      // pad_enable
                | (6u << 22)      // pad_interval: 128 dwords
                | (3u << 25));    // pad_amount:   4 dwords
  g1[1] = (int)((unsigned)DST << 16);   // tensor_dim0[15:0] = 128
  g1[2] = (int)((unsigned)CT << 16);    // tensor_dim1[15:0] = 64
  g1[3] = (int)((unsigned)DST << 16);   // tile_dim0 = 128
  g1[4] = (int)CT;                      // tile_dim1 = 64, tile_dim2 = 0
  g1[5] = (int)(NHEAD * DST);           // tensor_dim0_stride = 3072 elems
  g1[6] = 0;
  g1[7] = 0;
  i32x4 gz4 = {0, 0, 0, 0};
  i32x8 gz8 = {0, 0, 0, 0, 0, 0, 0, 0};
  // 6-arg toolchain variant: (g0, g1, g2, g3, g4, cpol)
  __builtin_amdgcn_tensor_load_to_lds(g0, g1, gz4, gz4, gz8, 0);
}
#endif

__global__ __launch_bounds__(NT, 1)
void ssm_ssd_wmma_kernel(const float* __restrict__ x,
                         const float* __restrict__ A,
                         const float* __restrict__ Bm,
                         const float* __restrict__ Cm,
                         const float* __restrict__ dparam,
                         float* __restrict__ out) {
  extern __shared__ float sm[];
  float* sX = sm + OFF_X;
  float* sH = sm + OFF_H;
  float* sS = sm + OFF_S;
  float* sLraw = sm + OFF_LRAW;
  float* sLg   = sm + OFF_LG;
  float* sE    = sm + OFF_E;
  float* sDT   = sm + OFF_DT;

  const int tid  = threadIdx.x;
  const int lane = tid & 31;
  const int wid  = tid >> 5;
  const int b = blockIdx.x / NHEAD;
  const int h = blockIdx.x % NHEAD;
  const float dp = dparam[h];

  for (int i = tid; i < SZ_H; i += NT) sH[i] = 0.0f;   // h0 = 0

#ifdef HAVE_TDM
  if (wid == 0) {  // preload chunk 0 B/C tiles via TDM
    size_t base0 = (((size_t)b * LL) * NHEAD + h) * DST;
    unsigned ldsB = (unsigned)(uintptr_t)(void*)(sm + OFF_B0) & 0xffffffffu;
    unsigned ldsC = (unsigned)(uintptr_t)(void*)(sm + OFF_C0) & 0xffffffffu;
    tdm_load_tile(Bm + base0, ldsB);
    tdm_load_tile(Cm + base0, ldsC);
  }
#endif
  __syncthreads();

#pragma unroll 1
  for (int c = 0; c < NC; ++c) {
    const int t0 = c * CT;
    const int buf = c & 1;
    float* sB = sm + (buf ? OFF_B1 : OFF_B0);
    float* sC = sm + (buf ? OFF_C1 : OFF_C0);

    // ---------------- stage X and A (regular loads) ----------------
    for (int idx = tid; idx < CT * (DHD / 4); idx += NT) {
      int t = idx >> 4, v = idx & 15;
      const float* rp = x + (((size_t)b * LL + t0 + t) * NHEAD + h) * DHD;
      reinterpret_cast<float4*>(&sX[t * XP])[v] =
          reinterpret_cast<const float4*>(rp)[v];
    }
    if (tid < CT) {
      float a = A[((size_t)b * LL + t0 + tid) * NHEAD + h];
      float z = a + dp;
      float delta = (z > 20.0f) ? z : log1pf(expf(z));   // softplus
      sLraw[tid] = -delta * fabsf(a);                    // log(a_bar)
    }
    if (c + 1 < NC && tid < CT) {
      __builtin_prefetch(
          &x[(((size_t)b * LL + t0 + CT + tid) * NHEAD + h) * DHD], 0, 0);
    }

#ifdef HAVE_TDM
    if (wid == 0) __builtin_amdgcn_s_wait_tensorcnt(0);  // chunk c B/C landed
    __syncthreads();
    if (wid == 0 && c + 1 < NC) {  // kick off chunk c+1 DMA into other buffer
      size_t nb = (((size_t)b * LL + t0 + CT) * NHEAD + h) * DST;
      unsigned ldsB =
          (unsigned)(uintptr_t)(void*)(sm + (buf ? OFF_B0 : OFF_B1));
      unsigned ldsC =
          (unsigned)(uintptr_t)(void*)(sm + (buf ? OFF_C0 : OFF_C1));
      tdm_load_tile(Bm + nb, ldsB);
      tdm_load_tile(Cm + nb, ldsC);
    }
#else
    for (int idx = tid; idx < CT * (DST / 4); idx += NT) {
      int t = idx >> 5, v = idx & 31;
      size_t base = (((size_t)b * LL + t0 + t) * NHEAD + h) * DST;
      reinterpret_cast<float4*>(&sB[t * BP])[v] =
          reinterpret_cast<const float4*>(Bm + base)[v];
      reinterpret_cast<float4*>(&sC[t * CP])[v] =
          reinterpret_cast<const float4*>(Cm + base)[v];
    }
    __syncthreads();
#endif

    // ---------------- prefix log-decay, scale tables ----------------
    if (tid < CT) {
      float acc = 0.0f, Lt = 0.0f;
      for (int k = 0; k < CT; ++k) {
        acc += sLraw[k];
        if (k == tid) Lt = acc;
      }
      sLg[tid] = Lt;
      sE[tid]  = expf(Lt);
      sDT[tid] = expf(acc - Lt);
    }
    __syncthreads();

    // -------- S = C_c * B_c^T (64x64, K=128): 2 tiles/wave, shared A --------
    {
      int mt = wid >> 1, ntb = (wid & 1) * 2;
      v8f acc0 = {}, acc1 = {};
#pragma unroll 8
      for (int k = 0; k < DST; k += 4) {
        v2f a  = ld_a(&sC[(mt * 16) * CP + k], CP);
        v2f b0 = ld_a(&sB[(ntb * 16) * BP + k], BP);        // B^T loads
        v2f b1 = ld_a(&sB[((ntb + 1) * 16) * BP + k], BP);
        acc0 = wmma_f32(a, b0, acc0);
        acc1 = wmma_f32(a, b1, acc1);
      }
      int n = lane & 15, mh = (lane >> 4) * 8;
#pragma unroll
      for (int r = 0; r < 8; ++r) {
        sS[(mt * 16 + mh + r) * SP + ntb * 16 + n] = acc0[r];
        sS[(mt * 16 + mh + r) * SP + (ntb + 1) * 16 + n] = acc1[r];
      }
    }
    __syncthreads();

    // ---------------- causal decay mask ----------------
    for (int i = tid; i < CT * CT; i += NT) {
      int t = i >> 6, u = i & 63;
      float g = sS[t * SP + u];
      sS[t * SP + u] = (u <= t) ? g * expf(sLg[t] - sLg[u]) : 0.0f;
    }
    __syncthreads();

    // ---- Y = G*X + diag(e^L)*C_c*Hprev^T : 2 tiles/wave, shared A ----
    {
      int mt = wid >> 1, ntb = (wid & 1) * 2;
      v8f acc0 = {}, acc1 = {};
#pragma unroll 8
      for (int k = 0; k < CT; k += 4) {                 // intra-chunk
        v2f a  = ld_a(&sS[(mt * 16) * SP + k], SP);
        v2f b0 = ld_b(&sX[k * XP + ntb * 16], XP);
        v2f b1 = ld_b(&sX[k * XP + (ntb + 1) * 16], XP);
        acc0 = wmma_f32(a, b0, acc0);
        acc1 = wmma_f32(a, b1, acc1);
      }
      float rs = sE[mt * 16 + (lane & 15)];             // row scale e^{L[t]}
#pragma unroll 8
      for (int k = 0; k < DST; k += 4) {                // carried state
        v2f a = ld_a(&sC[(mt * 16) * CP + k], CP);
        a.x *= rs; a.y *= rs;
        v2f b0 = ld_a(&sH[(ntb * 16) * HP + k], HP);    // Hprev^T loads
        v2f b1 = ld_a(&sH[((ntb + 1) * 16) * HP + k], HP);
        acc0 = wmma_f32(a, b0, acc0);
        acc1 = wmma_f32(a, b1, acc1);
      }
      int n = lane & 15, mh = (lane >> 4) * 8;
#pragma unroll
      for (int r = 0; r < 8; ++r) {
        int t = mt * 16 + mh + r;
        size_t o = (((size_t)b * LL + t0 + t) * NHEAD + h) * DHD;
        out[o + ntb * 16 + n] = acc0[r];
        out[o + (ntb + 1) * 16 + n] = acc1[r];
      }
    }
    __syncthreads();  // everyone done reading old sH

    // -- H = e^{Lend}*H + X^T*diag(e^{Lend-L})*B_c : 4 tiles/wave, shared A --
    {
      float Pd = sE[CT - 1];
      int mt = wid >> 1, ntb = (wid & 1) * 4;
      v8f a0 = {}, a1 = {}, a2 = {}, a3 = {};
      int k0 = (lane >> 4) << 1, m = lane & 15;
#pragma unroll 8
      for (int k = 0; k < CT; k += 4) {
        v2f a;                              // A[d][tau] = X[tau][d]*w[tau]
        a.x = sX[(k + k0) * XP + mt * 16 + m] * sDT[k + k0];
        a.y = sX[(k + k0 + 1) * XP + mt * 16 + m] * sDT[k + k0 + 1];
        v2f b0 = ld_b(&sB[k * BP + (ntb + 0) * 16], BP);
        v2f b1 = ld_b(&sB[k * BP + (ntb + 1) * 16], BP);
        v2f b2 = ld_b(&sB[k * BP + (ntb + 2) * 16], BP);
        v2f b3 = ld_b(&sB[k * BP + (ntb + 3) * 16], BP);
        a0 = wmma_f32(a, b0, a0);
        a1 = wmma_f32(a, b1, a1);
        a2 = wmma_f32(a, b2, a2);
        a3 = wmma_f32(a, b3, a3);
      }
      int n = lane & 15, mh = (lane >> 4) * 8;
#pragma unroll
      for (int r = 0; r < 8; ++r) {
        int d = mt * 16 + mh + r;
        float* hp = &sH[d * HP + n];
        hp[(ntb + 0) * 16] = Pd * hp[(ntb + 0) * 16] + a0[r];
        hp[(ntb + 1) * 16] = Pd * hp[(ntb + 1) * 16] + a1[r];
        hp[(ntb + 2) * 16] = Pd * hp[(ntb + 2) * 16] + a2[r];
        hp[(ntb + 3) * 16] = Pd * hp[(ntb + 3) * 16] + a3[r];
      }
    }
    __syncthreads();  // sX / B-C buffer reuse next chunk
  }
}

extern "C" void kernel_launch(void* const* d_in, const int* in_sizes, int n_in,
                              void* d_out, int out_size, void* d_ws, size_t ws_size,
                              hipStream_t stream) {
  const float* x  = (const float*)d_in[0];
  const float* A  = (const float*)d_in[1];
  const float* Bm = (const float*)d_in[2];
  const float* Cm = (const float*)d_in[3];
  const float* dp = (const float*)d_in[4];
  float* out = (float*)d_out;

  dim3 grid(BB * NHEAD);   // 48 independent (batch, head) scans
  dim3 block(NT);          // 8 wave32 waves
  size_t shmem = (size_t)LDS_FLOATS * sizeof(float);
  hipLaunchKernelGGL(ssm_ssd_wmma_kernel, grid, block, shmem, stream,
                     x, A, Bm, Cm, dp, out);
}